// GATEncoder_89215060672516
// MI455X (gfx1250) — compile-verified
//
#include <hip/hip_runtime.h>
#include <hip/hip_bf16.h>
#include <math.h>

typedef __attribute__((ext_vector_type(2))) float v2f;
typedef __attribute__((ext_vector_type(8))) float v8f;

#define HEADS 3
#define CH 64
#define OUTC 192
#define NEG_SLOPE 0.2f

__device__ __forceinline__ float lrelu(float x) { return x > 0.f ? x : NEG_SLOPE * x; }

// ---------------- init: zero accumulator, smax=-inf, denom=0 -----------------
__global__ void k_init(float* __restrict__ acc, float* __restrict__ smax,
                       float* __restrict__ denom, int n) {
    int i = blockIdx.x * blockDim.x + threadIdx.x;
    if (i < n * OUTC) acc[i] = 0.f;
    if (i < n * HEADS) { smax[i] = -3.0e38f; denom[i] = 0.f; }
}

// ---------------- dual GEMM: Yl = X @ Wl^T, Yr = X @ Wr^T --------------------
// X: [Nrows, K] row-major, Wl/Wr: [OUTC, K] row-major, Yl/Yr: [Nrows, OUTC].
// One wave computes one 16(row)x16(col) tile for BOTH outputs, sharing the
// A-fragment: 2x v_wmma_f32_16x16x4_f32 per K-step.
__global__ void k_gemm_dual(const float* __restrict__ X,
                            const float* __restrict__ Wl,
                            const float* __restrict__ Wr,
                            float* __restrict__ Yl, float* __restrict__ Yr,
                            int Nrows, int K) {
    int wid  = (blockIdx.x * blockDim.x + threadIdx.x) >> 5;
    int lane = threadIdx.x & 31;
    const int tiles_n = OUTC / 16;                 // 12
    int tile_m = wid / tiles_n;
    int tile_n = wid - tile_m * tiles_n;
    if (tile_m * 16 >= Nrows) return;              // wave-uniform

    int arow = tile_m * 16 + (lane & 15);          // A: M = lane%16
    int bcol = tile_n * 16 + (lane & 15);          // B: N = lane%16
    int koff = (lane < 16) ? 0 : 2;                // half-wave split of K

    const float* xrow  = X  + (size_t)arow * K;
    const float* wlrow = Wl + (size_t)bcol * K;
    const float* wrrow = Wr + (size_t)bcol * K;

    v8f cl = {}; v8f cr = {};
    for (int k = 0; k < K; k += 4) {
        v2f a, bl, br;
        a.x  = xrow[k + koff];      a.y  = xrow[k + koff + 1];
        bl.x = wlrow[k + koff];     bl.y = wlrow[k + koff + 1];
        br.x = wrrow[k + koff];     br.y = wrrow[k + koff + 1];
        cl = __builtin_amdgcn_wmma_f32_16x16x4_f32(false, a, false, bl,
                                                   (short)0, cl, false, false);
        cr = __builtin_amdgcn_wmma_f32_16x16x4_f32(false, a, false, br,
                                                   (short)0, cr, false, false);
    }
    // C/D layout: VGPR r -> row r (lanes 0-15) / row r+8 (lanes 16-31)
    int mbase = tile_m * 16 + ((lane < 16) ? 0 : 8);
    int col   = tile_n * 16 + (lane & 15);
#pragma unroll
    for (int r = 0; r < 8; ++r) {
        Yl[(size_t)(mbase + r) * OUTC + col] = cl[r];
        Yr[(size_t)(mbase + r) * OUTC + col] = cr[r];
    }
}

// ---------------- edge scores + segment max ----------------------------------
// One wave per edge; lanes sweep channels (c=lane, lane+32 inside each head).
__global__ void k_edge_score_max(const float* __restrict__ xl,
                                 const float* __restrict__ xr,
                                 const int* __restrict__ srcA,
                                 const int* __restrict__ dstA,
                                 const float* __restrict__ att,
                                 float* __restrict__ score,
                                 float* __restrict__ smax,
                                 int E, int n) {
    int wid  = (blockIdx.x * blockDim.x + threadIdx.x) >> 5;
    int lane = threadIdx.x & 31;
    int EP = E + n;
    if (wid >= EP) return;
    int s, d;
    if (wid < E) { s = srcA[wid]; d = dstA[wid]; } else { s = d = wid - E; }
    const float* pl = xl + (size_t)s * OUTC;
    const float* pr = xr + (size_t)d * OUTC;
#pragma unroll
    for (int h = 0; h < HEADS; ++h) {
        int c0 = h * CH + lane;
        float v = att[h * CH + lane]      * lrelu(pl[c0]      + pr[c0]) +
                  att[h * CH + lane + 32] * lrelu(pl[c0 + 32] + pr[c0 + 32]);
#pragma unroll
        for (int off = 16; off; off >>= 1) v += __shfl_xor(v, off, 32);
        if (lane == 0) {
            score[(size_t)wid * HEADS + h] = v;
            atomicMax(&smax[(size_t)d * HEADS + h], v);   // global_atomic_max_num_f32
        }
    }
}

// ---------------- softmax numerator + segment sum ----------------------------
__global__ void k_edge_exp_den(const int* __restrict__ dstA,
                               float* __restrict__ score,
                               const float* __restrict__ smax,
                               float* __restrict__ denom, int E, int n) {
    int i = blockIdx.x * blockDim.x + threadIdx.x;
    int EP = E + n;
    if (i >= EP * HEADS) return;
    int e = i / HEADS;
    int h = i - e * HEADS;
    int d = (e < E) ? dstA[e] : e - E;
    float w = __expf(score[i] - smax[d * HEADS + h]);
    score[i] = w;                                          // overwrite with numerator
    atomicAdd(&denom[d * HEADS + h], w);
}

// ---------------- weighted scatter aggregation -------------------------------
// One wave per edge; lanes cover the 192 channels (6 per lane), fully
// coalesced atomic adds into one destination row.
__global__ void k_edge_aggregate(const float* __restrict__ xl,
                                 const int* __restrict__ srcA,
                                 const int* __restrict__ dstA,
                                 const float* __restrict__ wbuf,
                                 const float* __restrict__ denom,
                                 float* __restrict__ acc, int E, int n) {
    int wid  = (blockIdx.x * blockDim.x + threadIdx.x) >> 5;
    int lane = threadIdx.x & 31;
    int EP = E + n;
    if (wid >= EP) return;
    int s, d;
    if (wid < E) { s = srcA[wid]; d = dstA[wid]; } else { s = d = wid - E; }
    const float* wrow = wbuf + (size_t)wid * HEADS;
    const float* drow = denom + (size_t)d * HEADS;
    float a0 = wrow[0] / drow[0];
    float a1 = wrow[1] / drow[1];
    float a2 = wrow[2] / drow[2];
    const float* pl = xl + (size_t)s * OUTC;
    float* pa = acc + (size_t)d * OUTC;
#pragma unroll
    for (int i = 0; i < 6; ++i) {
        int c = lane + 32 * i;
        float al = (i < 2) ? a0 : ((i < 4) ? a1 : a2);
        atomicAdd(&pa[c], pl[c] * al);                     // global_atomic_add_f32
    }
}

// ---------------- bias (+ optional relu) -------------------------------------
__global__ void k_bias_act(const float* __restrict__ acc,
                           const float* __restrict__ bias,
                           float* __restrict__ out, int n, int do_relu) {
    int i = blockIdx.x * blockDim.x + threadIdx.x;
    if (i >= n * OUTC) return;
    int c = i % OUTC;
    float v = acc[i] + bias[c];
    out[i] = do_relu ? fmaxf(v, 0.f) : v;
}

extern "C" void kernel_launch(void* const* d_in, const int* in_sizes, int n_in,
                              void* d_out, int out_size, void* d_ws, size_t ws_size,
                              hipStream_t stream) {
    const float* x    = (const float*)d_in[0];
    const int*   ei   = (const int*)  d_in[1];
    const float* Wl1  = (const float*)d_in[2];
    const float* Wr1  = (const float*)d_in[3];
    const float* att1 = (const float*)d_in[4];
    const float* b1   = (const float*)d_in[5];
    const float* Wl2  = (const float*)d_in[6];
    const float* Wr2  = (const float*)d_in[7];
    const float* att2 = (const float*)d_in[8];
    const float* b2   = (const float*)d_in[9];
    float* out = (float*)d_out;

    const int FIN = 128;
    int N = in_sizes[0] / FIN;
    int E = in_sizes[1] / 2;
    int EP = E + N;
    const int* srcA = ei;
    const int* dstA = ei + E;

    // workspace carve-up (all fp32)
    float* xl    = (float*)d_ws;                 // N*192
    float* xr    = xl    + (size_t)N * OUTC;     // N*192
    float* hbuf  = xr    + (size_t)N * OUTC;     // N*192  (layer-1 acc / hidden)
    float* score = hbuf  + (size_t)N * OUTC;     // EP*3
    float* smax  = score + (size_t)EP * HEADS;   // N*3
    float* denom = smax  + (size_t)N * HEADS;    // N*3

    int waves_gemm = ((N + 15) / 16) * (OUTC / 16);
    dim3 blk(256);
    int g_init  = (N * OUTC + 255) / 256;
    int g_gemm  = (waves_gemm + 7) / 8;          // 8 waves / 256-thread block
    int g_edgeW = (EP + 7) / 8;                  // wave-per-edge kernels
    int g_eh    = (EP * HEADS + 255) / 256;
    int g_elem  = (N * OUTC + 255) / 256;

    // ---------------- layer 1 ----------------
    k_init<<<g_init, blk, 0, stream>>>(hbuf, smax, denom, N);
    k_gemm_dual<<<g_gemm, blk, 0, stream>>>(x, Wl1, Wr1, xl, xr, N, FIN);
    k_edge_score_max<<<g_edgeW, blk, 0, stream>>>(xl, xr, srcA, dstA, att1,
                                                  score, smax, E, N);
    k_edge_exp_den<<<g_eh, blk, 0, stream>>>(dstA, score, smax, denom, E, N);
    k_edge_aggregate<<<g_edgeW, blk, 0, stream>>>(xl, srcA, dstA, score, denom,
                                                  hbuf, E, N);
    k_bias_act<<<g_elem, blk, 0, stream>>>(hbuf, b1, hbuf, N, 1);

    // ---------------- layer 2 ----------------
    k_init<<<g_init, blk, 0, stream>>>(out, smax, denom, N);
    k_gemm_dual<<<g_gemm, blk, 0, stream>>>(hbuf, Wl2, Wr2, xl, xr, N, OUTC);
    k_edge_score_max<<<g_edgeW, blk, 0, stream>>>(xl, xr, srcA, dstA, att2,
                                                  score, smax, E, N);
    k_edge_exp_den<<<g_eh, blk, 0, stream>>>(dstA, score, smax, denom, E, N);
    k_edge_aggregate<<<g_edgeW, blk, 0, stream>>>(xl, srcA, dstA, score, denom,
                                                  out, E, N);
    k_bias_act<<<g_elem, blk, 0, stream>>>(out, b2, out, N, 0);
}